// GroupKANLayer_20340965114038
// MI455X (gfx1250) — compile-verified
//
#include <hip/hip_runtime.h>
#include <hip/hip_bf16.h>

typedef __attribute__((ext_vector_type(2))) float v2f;
typedef __attribute__((ext_vector_type(8))) float v8f;

#define KAN_B   65536
#define KAN_G   8
#define KAN_F   32
#define KAN_NB  8
#define KAN_OD  16
#define KAN_C   (KAN_G * KAN_OD)      // 128 output channels
#define KAN_IN  (KAN_G * KAN_F)       // 256 input features
#define TILES_PER_BLOCK 8             // 16 rows per tile -> 128 rows per block
#define LDS_ROW_STRIDE 36             // floats; 36*r mod 64 -> distinct multiples of 4 (bank-conflict-free A reads)
#define LDS_GROUP_STRIDE (16 * LDS_ROW_STRIDE)

// ---------------------------------------------------------------------------
// Kernel 0: zero the 256-float stats accumulators (sums[128], sumsq[128])
// ---------------------------------------------------------------------------
__global__ void kan_zero_stats(float* stats) {
    stats[threadIdx.x] = 0.0f;
}

// ---------------------------------------------------------------------------
// Kernel 1: RBF activation + per-group GEMM via V_WMMA_F32_16X16X4_F32,
//           writes h[B,128] to workspace and accumulates per-channel
//           sum / sum-of-squares with fp32 atomics.
// One wave (32 lanes) == one group. Block = 8 waves = all 8 groups.
// ---------------------------------------------------------------------------
__global__ void __launch_bounds__(256)
kan_main(const float* __restrict__ x,
         const float* __restrict__ centres,      // [G,NB]
         const float* __restrict__ log_widths,   // [G,NB]
         const float* __restrict__ rbf_weights,  // [G,NB]
         const float* __restrict__ linear_weight,// [G]
         const float* __restrict__ Wp,           // [G,F,OD]
         const float* __restrict__ bp,           // [G,OD]
         float* __restrict__ h,                  // [B,128]
         float* __restrict__ sums,               // [128]
         float* __restrict__ sumsq)              // [128]
{
    __shared__ float lds[KAN_G * LDS_GROUP_STRIDE];

    const int lane = threadIdx.x & 31;
    const int g    = threadIdx.x >> 5;      // wave id == group id
    const int half = lane >> 4;             // 0: lanes 0-15, 1: lanes 16-31
    const int m16  = lane & 15;

    // --- RBF params (uniform per wave; broadcast loads, L2-resident) ---
    float cc[KAN_NB], isg[KAN_NB], rw[KAN_NB];
#pragma unroll
    for (int n = 0; n < KAN_NB; ++n) {
        cc[n]  = centres[g * KAN_NB + n];
        isg[n] = 1.0f / (__expf(log_widths[g * KAN_NB + n]) + 1e-6f);
        rw[n]  = rbf_weights[g * KAN_NB + n];
    }
    const float lw = linear_weight[g];

    // --- B fragments for all 8 K-steps (resident in 16 VGPRs) ---
    // 32-bit B 4x16 layout: col = lane%16; lanes 0-15 hold K=k0,k0+1 in
    // VGPR0/VGPR1, lanes 16-31 hold K=k0+2,k0+3.
    float bf0[8], bf1[8];
#pragma unroll
    for (int kk = 0; kk < 8; ++kk) {
        const int k0 = kk * 4 + 2 * half;
        bf0[kk] = Wp[g * (KAN_F * KAN_OD) + (k0    ) * KAN_OD + m16];
        bf1[kk] = Wp[g * (KAN_F * KAN_OD) + (k0 + 1) * KAN_OD + m16];
    }

    // --- bias (cancels in BN but kept for exact h semantics) ---
    const float bv = bp[g * KAN_OD + m16];
    const v8f bias = {bv, bv, bv, bv, bv, bv, bv, bv};

    float s0 = 0.0f, s1 = 0.0f;             // per-lane channel sum / sumsq
    float* myLds = &lds[g * LDS_GROUP_STRIDE];

    for (int t = 0; t < TILES_PER_BLOCK; ++t) {
        const long rowbase = (long)(blockIdx.x * TILES_PER_BLOCK + t) * 16;

        // activation: lane = feature index, 16 rows
#pragma unroll
        for (int r = 0; r < 16; ++r) {
            const float xv = x[(rowbase + r) * KAN_IN + g * KAN_F + lane];
            float a = lw * xv;
#pragma unroll
            for (int n = 0; n < KAN_NB; ++n) {
                const float z = (xv - cc[n]) * isg[n];
                a += rw[n] * __expf(-0.5f * z * z);
            }
            myLds[r * LDS_ROW_STRIDE + lane] = a;   // conflict-free store
        }
        __syncthreads();

        // 16x16 (rows x OD) tile = 8 chained fp32 WMMAs over K=32
        v8f acc = bias;
#pragma unroll
        for (int kk = 0; kk < 8; ++kk) {
            const int k0 = kk * 4 + 2 * half;
            v2f a, b;
            // A 16x4 layout: row = lane%16; VGPR0/1 = K = k0, k0+1
            a.x = myLds[m16 * LDS_ROW_STRIDE + k0];
            a.y = myLds[m16 * LDS_ROW_STRIDE + k0 + 1];
            b.x = bf0[kk];
            b.y = bf1[kk];
            acc = __builtin_amdgcn_wmma_f32_16x16x4_f32(
                false, a, false, b, (short)0, acc, false, false);
        }

        // D layout: VGPR r, lanes 0-15 -> row r, lanes 16-31 -> row r+8; col = lane%16
#pragma unroll
        for (int r = 0; r < 8; ++r) {
            const long row = rowbase + r + 8 * half;
            const float v = acc[r];
            h[row * KAN_C + g * KAN_OD + m16] = v;
            s0 += v;
            s1 += v * v;
        }
        __syncthreads();    // guard next tile's LDS writes vs this tile's reads
    }

    // lanes l and l+16 hold the same channel g*16 + (l%16)
    s0 += __shfl_down(s0, 16, 32);
    s1 += __shfl_down(s1, 16, 32);
    if (lane < 16) {
        atomicAdd(&sums [g * KAN_OD + lane], s0);
        atomicAdd(&sumsq[g * KAN_OD + lane], s1);
    }
}

// ---------------------------------------------------------------------------
// Kernel 2: finalize BatchNorm stats -> scale/shift per channel
// ---------------------------------------------------------------------------
__global__ void kan_stats(const float* __restrict__ sums,
                          const float* __restrict__ sumsq,
                          const float* __restrict__ gamma,
                          const float* __restrict__ beta,
                          float* __restrict__ scale,
                          float* __restrict__ shift)
{
    const int c = threadIdx.x;              // 128 threads
    const float invB = 1.0f / (float)KAN_B;
    const float mean = sums[c] * invB;
    const float var  = sumsq[c] * invB - mean * mean;   // biased, matches jnp.var
    const float sc   = gamma[c] * rsqrtf(var + 1e-5f);
    scale[c] = sc;
    shift[c] = beta[c] - mean * sc;
}

// ---------------------------------------------------------------------------
// Kernel 3: out = h * scale + shift  (float4 grid-stride; channel pattern is
// fixed per thread because the stride is a multiple of 128 floats)
// ---------------------------------------------------------------------------
__global__ void __launch_bounds__(256)
kan_norm(const float4* __restrict__ h4,
         const float* __restrict__ scale,
         const float* __restrict__ shift,
         float4* __restrict__ out4, int n4)
{
    const int tid = blockIdx.x * blockDim.x + threadIdx.x;
    const int c0  = (threadIdx.x * 4) & (KAN_C - 1);    // multiple of 4
    const float4 sc = *(const float4*)&scale[c0];
    const float4 sh = *(const float4*)&shift[c0];
    const int stride = gridDim.x * blockDim.x;
    for (int i = tid; i < n4; i += stride) {
        float4 v = h4[i];
        v.x = v.x * sc.x + sh.x;
        v.y = v.y * sc.y + sh.y;
        v.z = v.z * sc.z + sh.z;
        v.w = v.w * sc.w + sh.w;
        out4[i] = v;
    }
}

// ---------------------------------------------------------------------------
extern "C" void kernel_launch(void* const* d_in, const int* in_sizes, int n_in,
                              void* d_out, int out_size, void* d_ws, size_t ws_size,
                              hipStream_t stream) {
    const float* x             = (const float*)d_in[0];
    const float* centres       = (const float*)d_in[1];
    const float* log_widths    = (const float*)d_in[2];
    const float* rbf_weights   = (const float*)d_in[3];
    const float* linear_weight = (const float*)d_in[4];
    const float* Wp            = (const float*)d_in[5];
    const float* bp            = (const float*)d_in[6];
    const float* gamma         = (const float*)d_in[7];
    const float* beta          = (const float*)d_in[8];
    float* out = (float*)d_out;

    float* ws    = (float*)d_ws;
    float* h     = ws;                                   // B*128 floats = 32 MB
    float* sums  = ws + (size_t)KAN_B * KAN_C;           // 128
    float* sumsq = sums + KAN_C;                         // 128
    float* scale = sums + 2 * KAN_C;                     // 128
    float* shift = sums + 3 * KAN_C;                     // 128

    kan_zero_stats<<<1, 256, 0, stream>>>(sums);

    const int blocks = KAN_B / (16 * TILES_PER_BLOCK);   // 512
    kan_main<<<blocks, 256, 0, stream>>>(x, centres, log_widths, rbf_weights,
                                         linear_weight, Wp, bp, h, sums, sumsq);

    kan_stats<<<1, KAN_C, 0, stream>>>(sums, sumsq, gamma, beta, scale, shift);

    const int n4 = KAN_B * KAN_C / 4;
    kan_norm<<<1024, 256, 0, stream>>>((const float4*)h, scale, shift,
                                       (float4*)out, n4);
}